// IRCN_81209241632974
// MI455X (gfx1250) — compile-verified
//
#include <hip/hip_runtime.h>
#include <cstdint>
#include <cstddef>

// Instance+Row+Col normalization, [B,C,512,512] -> [B,3C,512,512].
// Memory-bound kernel: one workgroup per (b,c) plane, two passes.
// CDNA5 paths: double-buffered global_load_async_to_lds_b128 (ASYNCcnt),
// wave32 __shfl_xor reductions, ds_add_f32 LDS atomics, non-temporal B128 stores.

typedef float v4f __attribute__((ext_vector_type(4)));
typedef int   v4i __attribute__((vector_size(16)));   // matches builtin param type

static constexpr int   HDIM = 512;
static constexpr int   WDIM = 512;
static constexpr int   NTHREADS = 1024;   // 32 waves (wave32)
static constexpr int   NWAVES = 32;
static constexpr float kEPS = 1e-8f;

__device__ __forceinline__ void async_copy_b128(const float* gptr, float* lptr) {
#if __has_builtin(__builtin_amdgcn_global_load_async_to_lds_b128)
  __builtin_amdgcn_global_load_async_to_lds_b128(
      (__attribute__((address_space(1))) v4i*)(uintptr_t)gptr,
      (__attribute__((address_space(3))) v4i*)(uint32_t)(uintptr_t)lptr,
      0, 0);
#else
  uint32_t laddr = (uint32_t)(uintptr_t)lptr;   // low 32 bits of generic = LDS offset
  asm volatile("global_load_async_to_lds_b128 %0, %1, off"
               :
               : "v"(laddr), "v"(gptr)
               : "memory");
#endif
}

template <int N>
__device__ __forceinline__ void wait_async() {
  asm volatile("s_wait_asynccnt %0" ::"i"(N) : "memory");
}

__global__ __launch_bounds__(NTHREADS)
void ircn_kernel(const float* __restrict__ x,
                 const float* __restrict__ gRowP, const float* __restrict__ bRowP,
                 const float* __restrict__ gColP, const float* __restrict__ bColP,
                 const float* __restrict__ gInsP, const float* __restrict__ bInsP,
                 float* __restrict__ out, int C) {
  __shared__ __align__(16) float rowBuf[NWAVES * 2 * WDIM];  // 128 KB double-buffered staging
  __shared__ __align__(16) float rowMeanS[HDIM];
  __shared__ __align__(16) float rowRS[HDIM];
  __shared__ __align__(16) float rowSumS[HDIM];
  __shared__ __align__(16) float rowSum2S[HDIM];
  __shared__ __align__(16) float colMeanS[WDIM];  // sum -> mean
  __shared__ __align__(16) float colRS[WDIM];     // sumsq -> rstd
  __shared__ float insS[2];                       // mean, rstd

  const int tid  = threadIdx.x;
  const int wave = tid >> 5;
  const int lane = tid & 31;
  const int blk  = blockIdx.x;           // b*C + c
  const int c    = blk % C;
  const int b    = blk / C;

  const float* __restrict__ plane = x + (size_t)blk * (HDIM * WDIM);
  const size_t planeElems = (size_t)HDIM * WDIM;
  const size_t outBase = (size_t)b * 3 * C * planeElems;
  float* __restrict__ outIns = out + outBase + (size_t)c * planeElems;
  float* __restrict__ outRow = out + outBase + (size_t)(C + c) * planeElems;
  float* __restrict__ outCol = out + outBase + (size_t)(2 * C + c) * planeElems;

  if (tid < WDIM) { colMeanS[tid] = 0.f; colRS[tid] = 0.f; }
  __syncthreads();

  // ---------------- Phase A: statistics ----------------
  float* buf0 = rowBuf + (wave * 2) * WDIM;
  float* buf1 = buf0 + WDIM;
  const int laneOff = lane << 2;

  v4f cS[4]  = {};
  v4f cS2[4] = {};

  // Prime the pipeline: stage row (wave) into buf0.
  {
    const float* rowPtr = plane + (size_t)wave * WDIM;
#pragma unroll
    for (int j = 0; j < 4; ++j)
      async_copy_b128(rowPtr + laneOff + (j << 7), buf0 + laneOff + (j << 7));
  }

  for (int k = 0; k < 16; ++k) {
    const int h = wave + (k << 5);
    float* cur = (k & 1) ? buf1 : buf0;

    if (k < 15) {
      // Prefetch next row into the other buffer, then wait only for the
      // 4 oldest async ops (the current row) -- async loads complete in order.
      float* nxt = (k & 1) ? buf0 : buf1;
      const float* rowPtr = plane + (size_t)(h + 32) * WDIM;
#pragma unroll
      for (int j = 0; j < 4; ++j)
        async_copy_b128(rowPtr + laneOff + (j << 7), nxt + laneOff + (j << 7));
      wait_async<4>();
    } else {
      wait_async<0>();
    }

    float rS = 0.f, rS2 = 0.f;
#pragma unroll
    for (int j = 0; j < 4; ++j) {
      v4f v  = *(const v4f*)(cur + laneOff + (j << 7));
      v4f vv = v * v;
      rS  += v.x + v.y + v.z + v.w;
      rS2 += vv.x + vv.y + vv.z + vv.w;
      cS[j]  += v;
      cS2[j] += vv;
    }
    // wave32 reduction across lanes
#pragma unroll
    for (int m = 16; m >= 1; m >>= 1) {
      rS  += __shfl_xor(rS,  m, 32);
      rS2 += __shfl_xor(rS2, m, 32);
    }
    if (lane == 0) {
      const float mean = rS * (1.0f / WDIM);
      const float var  = fmaxf(rS2 * (1.0f / WDIM) - mean * mean, 0.f);
      rowMeanS[h] = mean;
      rowRS[h]    = 1.f / (sqrtf(var + kEPS) + kEPS);
      rowSumS[h]  = rS;
      rowSum2S[h] = rS2;
    }
  }

  // Merge per-lane column partials across waves (ds_add_f32).
#pragma unroll
  for (int j = 0; j < 4; ++j) {
    const int base = laneOff + (j << 7);
#pragma unroll
    for (int i = 0; i < 4; ++i) {
      atomicAdd(&colMeanS[base + i], cS[j][i]);
      atomicAdd(&colRS[base + i],   cS2[j][i]);
    }
  }
  __syncthreads();

  if (tid < WDIM) {  // waves 0..15 finalize column stats
    const float s = colMeanS[tid], s2 = colRS[tid];
    const float mean = s * (1.0f / HDIM);
    const float var  = fmaxf(s2 * (1.0f / HDIM) - mean * mean, 0.f);
    colMeanS[tid] = mean;
    colRS[tid]    = 1.f / (sqrtf(var + kEPS) + kEPS);
  }
  if (wave == 16) {  // instance stats from the 512 row sums
    float s = 0.f, s2 = 0.f;
#pragma unroll
    for (int k = 0; k < 16; ++k) {
      s  += rowSumS[lane + (k << 5)];
      s2 += rowSum2S[lane + (k << 5)];
    }
#pragma unroll
    for (int m = 16; m >= 1; m >>= 1) {
      s  += __shfl_xor(s,  m, 32);
      s2 += __shfl_xor(s2, m, 32);
    }
    if (lane == 0) {
      const float invN = 1.0f / (float)(HDIM * WDIM);
      const float mean = s * invN;
      const float var  = fmaxf(s2 * invN - mean * mean, 0.f);
      insS[0] = mean;
      insS[1] = 1.f / (sqrtf(var + kEPS) + kEPS);
    }
  }
  __syncthreads();

  // ---------------- Phase B: apply ----------------
  const float gR = gRowP[c], bR = bRowP[c];
  const float gC = gColP[c], bC = bColP[c];
  const float gI = gInsP[c], bI = bInsP[c];
  const float mI = insS[0];
  const float aI = gI * insS[1];

  for (int k = 0; k < 16; ++k) {
    const int h = wave + (k << 5);
    const size_t ro = (size_t)h * WDIM;
    const float rm = rowMeanS[h];
    const float aR = gR * rowRS[h];
#pragma unroll
    for (int j = 0; j < 4; ++j) {
      const int w0 = laneOff + (j << 7);
      v4f v  = *(const v4f*)(plane + ro + w0);
      v4f cm = *(const v4f*)&colMeanS[w0];
      v4f cr = *(const v4f*)&colRS[w0];

      v4f oI = (v - mI) * aI + bI;
      v4f oR = (v - rm) * aR + bR;
      v4f oC = (v - cm) * (cr * gC) + bC;

      __builtin_nontemporal_store(oI, (v4f*)(outIns + ro + w0));
      __builtin_nontemporal_store(oR, (v4f*)(outRow + ro + w0));
      __builtin_nontemporal_store(oC, (v4f*)(outCol + ro + w0));
    }
  }
}

extern "C" void kernel_launch(void* const* d_in, const int* in_sizes, int n_in,
                              void* d_out, int out_size, void* d_ws, size_t ws_size,
                              hipStream_t stream) {
  (void)d_ws; (void)ws_size; (void)out_size; (void)n_in;
  const float* x     = (const float*)d_in[0];
  const float* gRow  = (const float*)d_in[1];
  const float* bRow  = (const float*)d_in[2];
  const float* gCol  = (const float*)d_in[3];
  const float* bCol  = (const float*)d_in[4];
  const float* gIns  = (const float*)d_in[5];
  const float* bIns  = (const float*)d_in[6];

  const int C = in_sizes[1];                                   // 128
  const int B = in_sizes[0] / (C * HDIM * WDIM);               // 2

  dim3 grid(B * C);
  dim3 block(NTHREADS);
  ircn_kernel<<<grid, block, 0, stream>>>(x, gRow, bRow, gCol, bCol, gIns, bIns,
                                          (float*)d_out, C);
}